// HANLayer_47287589929193
// MI455X (gfx1250) — compile-verified
//
#include <hip/hip_runtime.h>

// ---------------------------------------------------------------------------
// HAN layer (3 x GCN + mean) for MI455X / gfx1250.
//
//  - Dense h@W done with V_WMMA_F32_16X16X4_F32 (exact fp32 path), operands
//    staged in LDS.  rsqrt(deg_out) folded into the GEMM epilogue.
//  - Edge scatter is wave-per-edge f32 atomic adds (L2-resident destination).
//  - Final epilogue fuses rsqrt(deg_in), bias, relu and the /3 mean.
// ---------------------------------------------------------------------------

typedef __attribute__((ext_vector_type(2))) float v2f;
typedef __attribute__((ext_vector_type(8))) float v8f;

#define IN_F  128
#define OUT_F 64

// ---------------- zero scratch ----------------
__global__ void zero_kernel(float* __restrict__ p, size_t n) {
    size_t i = (size_t)blockIdx.x * blockDim.x + threadIdx.x;
    if (i < n) p[i] = 0.0f;
}

// ---------------- degree histogram ----------------
__global__ void degree_kernel(const int* __restrict__ src, const int* __restrict__ dst,
                              float* __restrict__ degO, float* __restrict__ degI, int nE) {
    int e = blockIdx.x * blockDim.x + threadIdx.x;
    if (e < nE) {
        atomicAdd(degO + src[e], 1.0f);
        atomicAdd(degI + dst[e], 1.0f);
    }
}

// ---------------- deg -> rsqrt(max(deg,1)) in place ----------------
__global__ void rsqrt_kernel(float* __restrict__ degO, float* __restrict__ degI, int n) {
    int i = blockIdx.x * blockDim.x + threadIdx.x;
    if (i < n) {
        degO[i] = rsqrtf(fmaxf(degO[i], 1.0f));
        degI[i] = rsqrtf(fmaxf(degI[i], 1.0f));
    }
}

// ---------------- x = (h @ W) * rsqrt(deg_out)  via WMMA fp32 ----------------
// Block: 128 threads = 4 waves.  Block tile: 16 rows x 64 cols.
// Wave w computes the 16x16 tile at columns [16w, 16w+16).
// K loop: 32 iterations of V_WMMA_F32_16X16X4_F32.
//
// A (16x4 f32) VGPR layout (ISA 7.12.2): lanes 0-15 hold K=k+0 (v0), K=k+1 (v1)
// for row M=lane; lanes 16-31 hold K=k+2, K=k+3 for row M=lane-16.
// B (4x16) mirrored: lanes 0-15 hold B[k+0][n] (v0), B[k+1][n] (v1);
// lanes 16-31 hold B[k+2][n], B[k+3][n], n = lane&15.
// C/D: VGPR r = rows r (lanes 0-15) and r+8 (lanes 16-31), n = lane&15.
__global__ __launch_bounds__(128) void gemm_rs_kernel(const float* __restrict__ h,
                                                      const float* __restrict__ W,
                                                      const float* __restrict__ rsO,
                                                      float* __restrict__ x) {
    __shared__ float sW[IN_F * OUT_F];   // 32 KB
    __shared__ float sA[16 * IN_F];      //  8 KB

    const int tid  = threadIdx.x;
    const int row0 = blockIdx.x * 16;

    // Stage W (8192 floats) and the 16x128 h tile (2048 floats) into LDS.
    const float4* W4  = (const float4*)W;
    float4*       sW4 = (float4*)sW;
    #pragma unroll
    for (int i = tid; i < (IN_F * OUT_F) / 4; i += 128) sW4[i] = W4[i];

    const float4* h4  = (const float4*)(h + (size_t)row0 * IN_F);
    float4*       sA4 = (float4*)sA;
    #pragma unroll
    for (int i = tid; i < (16 * IN_F) / 4; i += 128) sA4[i] = h4[i];

    __syncthreads();

    const int wave = tid >> 5;
    const int lane = tid & 31;
    const int l16  = lane & 15;
    const int hi   = lane >> 4;          // 0: K pair (0,1)  1: K pair (2,3)
    const int col0 = wave << 4;

    const float* aRow = sA + l16 * IN_F + 2 * hi;
    const float* bCol = sW + (2 * hi) * OUT_F + col0 + l16;

    v8f c = {};
    #pragma unroll
    for (int k = 0; k < IN_F; k += 4) {
        v2f a, b;
        a.x = aRow[k];
        a.y = aRow[k + 1];
        b.x = bCol[k * OUT_F];
        b.y = bCol[k * OUT_F + OUT_F];
        // 8 args: (neg_a, A, neg_b, B, c_mod, C, reuse_a, reuse_b)
        c = __builtin_amdgcn_wmma_f32_16x16x4_f32(false, a, false, b,
                                                  (short)0, c, false, false);
    }

    #pragma unroll
    for (int r = 0; r < 8; ++r) {
        int row = row0 + r + hi * 8;
        x[(size_t)row * OUT_F + col0 + l16] = c[r] * rsO[row];
    }
}

// ---------------- agg[dst] += x[src]  (wave per edge, 2 feats/lane) ----------
__global__ __launch_bounds__(256) void spmm_kernel(const int* __restrict__ src,
                                                   const int* __restrict__ dst,
                                                   const float* __restrict__ x,
                                                   float* __restrict__ agg, int nE) {
    int e = blockIdx.x * 8 + (threadIdx.x >> 5);
    if (e >= nE) return;
    int lane = threadIdx.x & 31;
    int s = src[e];
    int d = dst[e];
    const float2 v = *(const float2*)(x + (size_t)s * OUT_F + lane * 2);
    float* a = agg + (size_t)d * OUT_F + lane * 2;
    atomicAdd(a + 0, v.x);
    atomicAdd(a + 1, v.y);
}

// ---------------- out (+)= relu(agg*rsqrt(deg_in) + b) / 3 -------------------
__global__ void combine_kernel(const float* __restrict__ agg,
                               const float* __restrict__ rsI,
                               const float* __restrict__ b,
                               float* __restrict__ out, int first, size_t n) {
    size_t i = (size_t)blockIdx.x * blockDim.x + threadIdx.x;
    if (i < n) {
        size_t node = i >> 6;     // OUT_F == 64
        int    f    = (int)(i & 63);
        float v = fmaxf(fmaf(agg[i], rsI[node], b[f]), 0.0f) * (1.0f / 3.0f);
        out[i] = first ? v : (out[i] + v);
    }
}

// ---------------------------------------------------------------------------
extern "C" void kernel_launch(void* const* d_in, const int* in_sizes, int n_in,
                              void* d_out, int out_size, void* d_ws, size_t ws_size,
                              hipStream_t stream) {
    (void)n_in; (void)out_size; (void)ws_size;

    const float* h   = (const float*)d_in[0];
    float*       out = (float*)d_out;
    float*       ws  = (float*)d_ws;

    const int N = in_sizes[0] / IN_F;    // 50000 (divisible by 16)

    // Scratch layout (floats): degO[N] | degI[N] | agg[64N] | x[64N]
    float* degO = ws;
    float* degI = ws + N;
    float* agg  = ws + 2 * (size_t)N;
    float* xbuf = ws + 2 * (size_t)N + (size_t)N * OUT_F;

    const size_t zeroN  = (size_t)N * (2 + OUT_F);   // degO+degI+agg
    const size_t elemsN = (size_t)N * OUT_F;

    for (int i = 0; i < 3; ++i) {
        const int*   src = (const int*)d_in[1 + 4 * i];
        const int*   dst = (const int*)d_in[2 + 4 * i];
        const float* W   = (const float*)d_in[3 + 4 * i];
        const float* b   = (const float*)d_in[4 + 4 * i];
        const int    E   = in_sizes[1 + 4 * i];

        zero_kernel<<<(unsigned)((zeroN + 255) / 256), 256, 0, stream>>>(ws, zeroN);
        degree_kernel<<<(E + 255) / 256, 256, 0, stream>>>(src, dst, degO, degI, E);
        rsqrt_kernel<<<(N + 255) / 256, 256, 0, stream>>>(degO, degI, N);
        gemm_rs_kernel<<<N / 16, 128, 0, stream>>>(h, W, degO, xbuf);
        spmm_kernel<<<(E + 7) / 8, 256, 0, stream>>>(src, dst, xbuf, agg, E);
        combine_kernel<<<(unsigned)((elemsN + 255) / 256), 256, 0, stream>>>(
            agg, degI, b, out, i == 0 ? 1 : 0, elemsN);
    }
}